// BackwardConv2D_8950711846021
// MI455X (gfx1250) — compile-verified
//
#include <hip/hip_runtime.h>

// CROWN backward through 3x3 SAME conv, B=4, H=W=16, CIN=COUT=8.
// Outputs (concatenated): w_out_u_ (4x2048x2048), b_out_u_ (4x2048),
//                         w_out_l_ (4x2048x2048), b_out_l_ (4x2048)
//
// Identity: max(w,0)+min(w,0) == w, so
//   w_out_*_ [b] = Wm @ w_out_*[b]      (Wm = banded dense conv matrix)
//   b_out_*_ [b] = b_out_*[b] + w_out_*[b]^T @ bvec,  bvec[o] = bias[o%8]
//
// A 16-row output tile (2 adjacent spatial cols x 8 ci) contracts over only a
// 3x4 spatial window x 8 co = K=96. SAME-padding zeros are folded into the A
// operand (zero A-column == zero B-row), so B loads are unconditional from
// clamped rows: no branches, pure scale_offset global loads. f32 WMMA
// 16x16x4 keeps exact reference precision; ~270MB HBM @ 23.3TB/s ~= 12us,
// 192MB L2 absorbs the 9x spatial reuse of w_out.

typedef __attribute__((ext_vector_type(2))) float v2f;
typedef __attribute__((ext_vector_type(8))) float v8f;

#define NK 2048   // n_back (GEMM N)
#define NO 2048   // n_out
#define HH 16
#define WW 16

__device__ __forceinline__ int clampi(int x, int lo, int hi) {
    return x < lo ? lo : (x > hi ? hi : x);   // -> v_med3_i32
}

__global__ __launch_bounds__(256) void crown_conv_back_gemm(
    const float* __restrict__ kern,   // (3,3,8,8) HWIO
    const float* __restrict__ wu,     // (4,1,2048,2048)
    const float* __restrict__ wl,
    float* __restrict__ out_u,
    float* __restrict__ out_l)
{
    // Per-block tables: A tile (mask folded in) + clamped B-row offsets.
    __shared__ float    Atab[16 * 96];   // [M][kk]
    __shared__ unsigned offtab[96];      // clamped row offset (elements)

    const int tid = threadIdx.x;
    const int rt  = blockIdx.x;           // row tile 0..127 (16 rows each)
    const int hi  = (2 * rt) >> 4;        // spatial row of the output pair
    const int wi0 = (2 * rt) & 15;        // even spatial col (pair = wi0,wi0+1)

    // ---- Build A (16 x 96): Atab[M][kk], M = ds*8+ci, kk = (h''*4+w'')*8+co.
    // Wm[(hi,wi,ci),(ho,wo,co)] = kernel[hi-ho+1, wi-wo+1, ci, co].
    // Zero columns for taps outside the kernel OR outside the image (padding).
    for (int idx = tid; idx < 16 * 96; idx += 256) {
        const int M   = idx / 96;
        const int kk  = idx - M * 96;
        const int ds  = M >> 3;
        const int ci  = M & 7;
        const int co  = kk & 7;
        const int wpp = (kk >> 3) & 3;
        const int hpp = kk >> 5;
        const int kh  = 2 - hpp;
        const int kw  = 2 + ds - wpp;
        const int ho  = hi  - 1 + hpp;
        const int wo  = wi0 - 1 + wpp;
        const bool valid = (kw >= 0) && (kw <= 2) &&
                           (ho >= 0) && (ho < HH) && (wo >= 0) && (wo < WW);
        Atab[idx] = valid ? kern[(kh * 3 + kw) * 64 + ci * 8 + co] : 0.0f;
    }
    // ---- Clamped B-row offsets (invalid rows read in-bounds garbage that is
    // multiplied by the zero A column).
    for (int kk = tid; kk < 96; kk += 256) {
        const int co  = kk & 7;
        const int wpp = (kk >> 3) & 3;
        const int hpp = kk >> 5;
        const int ho  = clampi(hi  - 1 + hpp, 0, HH - 1);
        const int wo  = clampi(wi0 - 1 + wpp, 0, WW - 1);
        offtab[kk] = (unsigned)(((ho * WW + wo) * 8 + co) << 11);  // * NK
    }
    __syncthreads();

    const int lane = tid & 31;
    const int wave = tid >> 5;            // 8 waves per block (wave32)
    const int mrow = lane & 15;           // A-row / B-col / D-col in tile
    const int half = lane >> 4;           // lanes 0-15 vs 16-31

    // batch/branch selection
    const int m      = blockIdx.z;        // 0..7
    const int b      = m & 3;
    const int branch = m >> 2;            // 0 = upper, 1 = lower
    const float* __restrict__ Win = (branch ? wl : wu) + (size_t)b * NO * NK;
    float* __restrict__ Out       = (branch ? out_l : out_u) + (size_t)b * NO * NK;

    const int k0 = blockIdx.y * 512 + wave * 64;   // this wave's 64 k-columns

    // ---- Pull A into registers: 24 x ds_load_b64.
    // A layout (16x4 f32): VGPR e, lanes0-15 -> K=4t+e ; lanes16-31 -> K=4t+e+2.
    const float* Arow = &Atab[mrow * 96 + 2 * half];
    v2f A[24];
    #pragma unroll
    for (int t = 0; t < 24; ++t)
        A[t] = *reinterpret_cast<const v2f*>(Arow + 4 * t);

    v8f zero = {};
    v8f acc[4];
    #pragma unroll
    for (int j = 0; j < 4; ++j) acc[j] = zero;

    const unsigned cb = (unsigned)(k0 + mrow);   // per-lane column base

    // ---- 24 K-chunks x 4 col-tiles: branchless gathered B, f32 WMMA.
    #pragma unroll
    for (int t = 0; t < 24; ++t) {
        const uint2 offp =
            *reinterpret_cast<const uint2*>(&offtab[4 * t + 2 * half]);
        const unsigned i0 = offp.x + cb;   // row for K=4t+{0|2}
        const unsigned i1 = offp.y + cb;   // row for K=4t+{1|3}
        #pragma unroll
        for (int j = 0; j < 4; ++j) {
            v2f Bv;
            Bv[0] = Win[i0 + 16u * j];     // saddr + voffset, imm offset 64*j
            Bv[1] = Win[i1 + 16u * j];
            acc[j] = __builtin_amdgcn_wmma_f32_16x16x4_f32(
                false, A[t], false, Bv, (short)0, acc[j], false, false);
        }
    }

    // ---- Store D: VGPR v, lanes0-15 -> M=v ; lanes16-31 -> M=v+8.
    const unsigned rowb = (unsigned)(rt * 16);
    #pragma unroll
    for (int v = 0; v < 8; ++v) {
        const unsigned Mr = (unsigned)(v + 8 * half);
        const unsigned rbase = (rowb + Mr) * (unsigned)NK + cb;
        #pragma unroll
        for (int j = 0; j < 4; ++j) {
            Out[rbase + 16u * j] = acc[j][v];
        }
    }
}

// b_out_*_ [b,k] = b_out_*[b,k] + sum_o w_out_*[b,o,k] * bias[o & 7]
__global__ __launch_bounds__(256) void crown_conv_back_bias(
    const float* __restrict__ bias,
    const float* __restrict__ wu, const float* __restrict__ wl,
    const float* __restrict__ bu, const float* __restrict__ bl,
    float* __restrict__ outbu, float* __restrict__ outbl)
{
    const int m      = blockIdx.y;
    const int b      = m & 3;
    const int branch = m >> 2;
    const float* __restrict__ Win = (branch ? wl : wu) + (size_t)b * NO * NK;
    const float* __restrict__ Bin = (branch ? bl : bu) + (size_t)b * NK;
    float* __restrict__ Bout      = (branch ? outbl : outbu) + (size_t)b * NK;

    const unsigned k = blockIdx.x * 256u + threadIdx.x;

    float br[8];
    #pragma unroll
    for (int c = 0; c < 8; ++c) br[c] = bias[c];   // uniform -> scalar loads

    float acc = Bin[k];
    for (unsigned o = 0; o < NO; o += 8) {
        #pragma unroll
        for (unsigned c = 0; c < 8; ++c)
            acc += Win[(o + c) * (unsigned)NK + k] * br[c];
    }
    Bout[k] = acc;
}

extern "C" void kernel_launch(void* const* d_in, const int* in_sizes, int n_in,
                              void* d_out, int out_size, void* d_ws, size_t ws_size,
                              hipStream_t stream) {
    // setup_inputs order: x(0), kernel(1), bias(2), w_out_u(3), b_out_u(4),
    //                     w_out_l(5), b_out_l(6)
    const float* kern = (const float*)d_in[1];
    const float* bias = (const float*)d_in[2];
    const float* wu   = (const float*)d_in[3];
    const float* bu   = (const float*)d_in[4];
    const float* wl   = (const float*)d_in[5];
    const float* bl   = (const float*)d_in[6];

    float* out    = (float*)d_out;
    float* out_wu = out;                                   // 4*2048*2048
    float* out_bu = out + 16777216;                        // 4*2048
    float* out_wl = out + 16777216 + 8192;                 // 4*2048*2048
    float* out_bl = out + 16777216 + 8192 + 16777216;      // 4*2048

    dim3 grid(128, 4, 8);   // 128 row tiles x 4 k-groups x (4 batch * 2 branch)
    crown_conv_back_gemm<<<grid, 256, 0, stream>>>(kern, wu, wl, out_wu, out_wl);
    crown_conv_back_bias<<<dim3(8, 8), 256, 0, stream>>>(bias, wu, wl, bu, bl,
                                                         out_bu, out_bl);
}